// ModelNew_70918499991670
// MI455X (gfx1250) — compile-verified
//
#include <hip/hip_runtime.h>

typedef float v2f __attribute__((ext_vector_type(2)));
typedef float v4f __attribute__((ext_vector_type(4)));
typedef float v8f __attribute__((ext_vector_type(8)));
typedef unsigned int u32x4 __attribute__((ext_vector_type(4)));
typedef int i32x4 __attribute__((ext_vector_type(4)));
typedef int i32x8 __attribute__((ext_vector_type(8)));

#if defined(__has_builtin)
#if __has_builtin(__builtin_amdgcn_tensor_load_to_lds) && __has_builtin(__builtin_amdgcn_s_wait_tensorcnt)
#define KDA_HAVE_TDM 1
#endif
#endif
#ifndef KDA_HAVE_TDM
#define KDA_HAVE_TDM 0
#endif

namespace {
constexpr int Bn = 4, Tn = 2048, Hn = 16, Kd = 128, Vd = 128;
constexpr int Cn  = 64;            // chunk length
constexpr int NC  = Tn / Cn;       // 32 chunks, sequential per (b,h)
constexpr int TD  = 132;           // LDS row stride, 128-wide tiles (+4 pad vs 64 banks)
constexpr int SD  = 132;           // LDS row stride, state
constexpr int LD2 = 68;            // LDS row stride, 64-wide (L / Attn) (+4 pad)
constexpr int SMEM_FLOATS = Kd*SD + 5*Cn*TD + 2*Cn*LD2 + Kd + Cn;
}

// D = A(16x4, f32) * B(4x16, f32) + C(16x16, f32)  — native CDNA5 f32 WMMA
__device__ __forceinline__ v8f wmma4(v2f a, v2f b, v8f c) {
  return __builtin_amdgcn_wmma_f32_16x16x4_f32(false, a, false, b, (short)0, c,
                                               false, false);
}

#if KDA_HAVE_TDM
// Issue a TDM DMA: 2D tile (rows x cols f32) from global into LDS, with the
// D# pad feature inserting 4 DWORDs of padding every 128 DWORDs so rows land
// at the kernel's TD=132 stride.  Fields per CDNA5 ISA.
__device__ __forceinline__ void tdm_load_2d(unsigned lds_off, const float* gptr,
                                            unsigned rows, unsigned cols,
                                            unsigned row_stride_elems) {
  unsigned long long ga = (unsigned long long)(uintptr_t)gptr;
  u32x4 g0;
  g0[0] = 1u;                                              // count=1 (valid), user mode
  g0[1] = lds_off;                                         // LDS byte address
  g0[2] = (unsigned)ga;                                    // global_addr[31:0]
  g0[3] = (unsigned)((ga >> 32) & 0x01FFFFFFu) | (2u << 30); // global_addr[56:32] | type=2
  i32x8 g1;
  // data_size=4B (2<<16) | pad_enable (1<<20) | pad_interval=128 DW (6<<22)
  // | pad_amount=4 DW (3<<25); workgroup_mask=0 (not in a cluster)
  g1[0] = (int)((2u << 16) | (1u << 20) | (6u << 22) | (3u << 25));
  g1[1] = (int)((cols & 0xFFFFu) << 16);                   // tensor_dim0[15:0] @ [63:48]
  g1[2] = (int)(((cols >> 16) & 0xFFFFu) | ((rows & 0xFFFFu) << 16)); // dim0 hi | dim1 lo
  g1[3] = (int)(((rows >> 16) & 0xFFFFu) | ((cols & 0xFFFFu) << 16)); // dim1 hi | tile_dim0
  g1[4] = (int)(rows & 0xFFFFu);                           // tile_dim1 | tile_dim2=0
  g1[5] = (int)row_stride_elems;                           // tensor_dim0_stride[31:0]
  g1[6] = 0;                                               // stride0 hi | stride1 lo
  g1[7] = 0;                                               // stride1 hi
  i32x4 z4 = {0, 0, 0, 0};
#if __clang_major__ >= 23
  i32x8 z8 = {0, 0, 0, 0, 0, 0, 0, 0};
  __builtin_amdgcn_tensor_load_to_lds(g0, g1, z4, z4, z8, 0);
#else
  __builtin_amdgcn_tensor_load_to_lds(g0, g1, z4, z4, 0);
#endif
}
#endif

__global__ __launch_bounds__(256)
void kda_chunked_kernel(const float* __restrict__ q, const float* __restrict__ k,
                        const float* __restrict__ v, const float* __restrict__ g,
                        const float* __restrict__ beta, float* __restrict__ o) {
  extern __shared__ float sm[];
  float* S     = sm;                 // [128][SD]  running state, LDS-resident
  float* Ktil  = S    + Kd * SD;     // [64][TD]   k * exp(cumg)
  float* Khat  = Ktil + Cn * TD;     // [64][TD]   k / exp(cumg)
  float* Qtil  = Khat + Cn * TD;     // [64][TD]   q * exp(cumg) * scale
  float* WD    = Qtil + Cn * TD;     // [64][TD]   g -> W -> Delta
  float* Vbuf  = WD   + Cn * TD;     // [64][TD]   v chunk (TDM destination)
  float* Lm    = Vbuf + Cn * TD;     // [64][LD2]  strictly-lower K~K^^T
  float* Am    = Lm   + Cn * LD2;    // [64][LD2]  incl-lower Q~K^^T
  float* eC    = Am   + Cn * LD2;    // [128]      exp(chunk-total decay)
  float* betaS = eC   + Kd;          // [64]

  const int tid  = threadIdx.x;
  const int wv   = tid >> 5;
  const int lane = tid & 31;
  const int ml   = lane & 15;        // tile M (A/C rows) or N (B/C cols)
  const int hi   = lane >> 4;        // half-wave selector
  const int bh   = blockIdx.x;
  const int b    = bh / Hn, h = bh % Hn;
  const float scale = 0.08838834764831845f;   // K^-0.5

  for (int i = tid; i < Kd * SD; i += 256) S[i] = 0.f;
  __syncthreads();

  const size_t strideT = (size_t)Hn * Kd;     // elements per timestep row

  for (int c = 0; c < NC; ++c) {
    const int t0 = c * Cn;
    const size_t base = (((size_t)b * Tn + t0) * Hn + h) * (size_t)Kd;

    // ---- stage q,k,g,v chunks (64x128 f32 tiles) into LDS ----
#if KDA_HAVE_TDM
    if (wv == 0) {   // one wave drives the Tensor Data Mover; TENSORcnt tracks
      const unsigned qOff = (unsigned)((char*)Qtil - (char*)sm);
      const unsigned kOff = (unsigned)((char*)Ktil - (char*)sm);
      const unsigned gOff = (unsigned)((char*)WD   - (char*)sm);
      const unsigned vOff = (unsigned)((char*)Vbuf - (char*)sm);
      tdm_load_2d(qOff, q + base, Cn, Kd, (unsigned)strideT);
      tdm_load_2d(kOff, k + base, Cn, Kd, (unsigned)strideT);
      tdm_load_2d(gOff, g + base, Cn, Kd, (unsigned)strideT);
      tdm_load_2d(vOff, v + base, Cn, Kd, (unsigned)strideT);
      __builtin_amdgcn_s_wait_tensorcnt(0);
    }
#else
    for (int i = 0; i < 8; ++i) {
      int f4 = tid + 256 * i;
      int r = f4 >> 5, c4 = (f4 & 31) << 2;
      size_t gi = base + (size_t)r * strideT + c4;
      *(v4f*)(Qtil + r * TD + c4) = *(const v4f*)(q + gi);
      *(v4f*)(Ktil + r * TD + c4) = *(const v4f*)(k + gi);
      *(v4f*)(WD   + r * TD + c4) = *(const v4f*)(g + gi);
      *(v4f*)(Vbuf + r * TD + c4) = *(const v4f*)(v + gi);
    }
#endif
    if (tid < Cn) betaS[tid] = beta[((size_t)b * Tn + t0 + tid) * Hn + h];
    __syncthreads();

    // ---- per-key cumulative decay; build K~, K^, Q~ (thread = key column) ----
    if (tid < Kd) {
      float acc = 0.f;
      for (int t = 0; t < Cn; ++t) {
        acc += WD[t * TD + tid];                 // g cumsum (g <= 0)
        float e  = __expf(acc);
        float kv = Ktil[t * TD + tid];
        Ktil[t * TD + tid] = kv * e;
        Khat[t * TD + tid] = kv / e;
        Qtil[t * TD + tid] = Qtil[t * TD + tid] * e * scale;
        if (t == Cn - 1) eC[tid] = e;
      }
    }
    __syncthreads();

    // ---- M1: Lm = tril<(K~ K^T),  Am = tril<=(Q~ K^T)   (64x64, inner 128) ----
    {
      const float* Asrc = (wv < 4) ? Ktil : Qtil;
      float* Dst = (wv < 4) ? Lm : Am;
      const int tr = wv & 3;
      const int m  = tr * 16 + ml;
      v8f acc[4] = {};
      for (int kb = 0; kb < Kd; kb += 4) {
        int k0 = kb + 2 * hi;
        v2f a; a.x = Asrc[m * TD + k0]; a.y = Asrc[m * TD + k0 + 1];
        for (int tc = 0; tc < 4; ++tc) {
          int n = tc * 16 + ml;                  // B[k][n] = Khat[n][k] (K^ transposed)
          v2f bf; bf.x = Khat[n * TD + k0]; bf.y = Khat[n * TD + k0 + 1];
          acc[tc] = wmma4(a, bf, acc[tc]);
        }
      }
      for (int tc = 0; tc < 4; ++tc)
        for (int j = 0; j < 8; ++j) {
          int tt = tr * 16 + j + 8 * hi;
          int ss = tc * 16 + ml;
          bool keep = (wv < 4) ? (ss < tt) : (ss <= tt);
          Dst[tt * LD2 + ss] = keep ? acc[tc][j] : 0.f;
        }
    }

    // ---- M2: W = V - K~ S0 (into WD) and O-part accO = Q~ S0 (regs) ----
    const int tr2 = wv >> 1, tc0 = (wv & 1) * 4;
    v8f accO[4] = {};
    {
      v8f accW[4] = {};
      const int m = tr2 * 16 + ml;
      for (int kb = 0; kb < Kd; kb += 4) {
        int k0 = kb + 2 * hi;
        v2f aK; aK.x = Ktil[m * TD + k0]; aK.y = Ktil[m * TD + k0 + 1];
        v2f aQ; aQ.x = Qtil[m * TD + k0]; aQ.y = Qtil[m * TD + k0 + 1];
        for (int t = 0; t < 4; ++t) {
          int n = (tc0 + t) * 16 + ml;
          v2f bf; bf.x = S[k0 * SD + n]; bf.y = S[(k0 + 1) * SD + n];
          accW[t] = wmma4(aK, bf, accW[t]);
          accO[t] = wmma4(aQ, bf, accO[t]);
        }
      }
      for (int t = 0; t < 4; ++t)
        for (int j = 0; j < 8; ++j) {
          int tt = tr2 * 16 + j + 8 * hi;
          int vv = (tc0 + t) * 16 + ml;
          WD[tt * TD + vv] = Vbuf[tt * TD + vv] - accW[t][j];  // unique owner
        }
    }
    __syncthreads();

    // ---- forward substitution: Delta = (I + diag(beta) L)^-1 diag(beta) W ----
    for (int s = 0; s < Cn; ++s) {
      if (tid < Vd) WD[s * TD + tid] *= betaS[s];    // finalize delta_s
      __syncthreads();
      int rem = Cn - 1 - s;
      for (int idx = tid; idx < rem * Vd; idx += 256) {
        int t  = s + 1 + (idx >> 7);
        int vv = idx & 127;
        WD[t * TD + vv] -= Lm[t * LD2 + s] * WD[s * TD + vv];
      }
      __syncthreads();
    }

    // ---- M3: O = accO + Am @ Delta (inner 64); store to global ----
    {
      const int m = tr2 * 16 + ml;
      for (int kb = 0; kb < Cn; kb += 4) {
        int k0 = kb + 2 * hi;
        v2f a; a.x = Am[m * LD2 + k0]; a.y = Am[m * LD2 + k0 + 1];
        for (int t = 0; t < 4; ++t) {
          int n = (tc0 + t) * 16 + ml;
          v2f bf; bf.x = WD[k0 * TD + n]; bf.y = WD[(k0 + 1) * TD + n];
          accO[t] = wmma4(a, bf, accO[t]);
        }
      }
      for (int t = 0; t < 4; ++t)
        for (int j = 0; j < 8; ++j) {
          int tt = tr2 * 16 + j + 8 * hi;
          int vv = (tc0 + t) * 16 + ml;
          o[(((size_t)b * Tn + t0 + tt) * Hn + h) * Vd + vv] = accO[t][j];
        }
    }

    // ---- M4: S = diag(eC) S + (K^ * eC)^T @ Delta  (128x128, inner 64) ----
    {
      const int kr = wv;                 // each wave owns 16 state rows
      const int mrow = kr * 16 + ml;     // key index for A rows
      const float eA = eC[mrow];
      v8f accS[8];
      for (int t = 0; t < 8; ++t)
        for (int j = 0; j < 8; ++j) {
          int kk = kr * 16 + j + 8 * hi;
          int vv = t * 16 + ml;
          accS[t][j] = eC[kk] * S[kk * SD + vv];
        }
      for (int kb = 0; kb < Cn; kb += 4) {
        int k0 = kb + 2 * hi;            // time index s
        v2f a; a.x = Khat[k0 * TD + mrow] * eA;
               a.y = Khat[(k0 + 1) * TD + mrow] * eA;
        for (int t = 0; t < 8; ++t) {
          int n = t * 16 + ml;
          v2f bf; bf.x = WD[k0 * TD + n]; bf.y = WD[(k0 + 1) * TD + n];
          accS[t] = wmma4(a, bf, accS[t]);
        }
      }
      for (int t = 0; t < 8; ++t)
        for (int j = 0; j < 8; ++j) {
          int kk = kr * 16 + j + 8 * hi;
          int vv = t * 16 + ml;
          S[kk * SD + vv] = accS[t][j];
        }
    }
    __syncthreads();
  }
}

extern "C" void kernel_launch(void* const* d_in, const int* in_sizes, int n_in,
                              void* d_out, int out_size, void* d_ws, size_t ws_size,
                              hipStream_t stream) {
  (void)in_sizes; (void)n_in; (void)out_size; (void)d_ws; (void)ws_size;
  const float* q    = (const float*)d_in[0];
  const float* k    = (const float*)d_in[1];
  const float* v    = (const float*)d_in[2];
  const float* g    = (const float*)d_in[3];
  const float* beta = (const float*)d_in[4];
  float* o = (float*)d_out;

  const size_t smem = (size_t)SMEM_FLOATS * sizeof(float);   // ~266 KB / WGP
  (void)hipFuncSetAttribute((const void*)kda_chunked_kernel,
                            hipFuncAttributeMaxDynamicSharedMemorySize, (int)smem);
  dim3 grid(Bn * Hn);   // 64 workgroups, one per (b,h)
  dim3 block(256);      // 8 wave32
  kda_chunked_kernel<<<grid, block, smem, stream>>>(q, k, v, g, beta, o);
}